// EdgeUnet_45286135169793
// MI455X (gfx1250) — compile-verified
//
#include <hip/hip_runtime.h>
#include <hip/hip_bf16.h>

// ---------------------------------------------------------------------------
// Graph U-Net (EdgeConv + TopK pooling) for MI455X / gfx1250, wave32.
// Dense GEMMs use V_WMMA_F32_16X16X4_F32 (one 16x16 tile per wave) with a
// mask-free inner loop: A via clamped-row b64 loads * 0/1 mask, B pre-packed
// into WMMA fragment order so each lane does one coalesced b64 load per
// K-step. Segment-max uses ordered-uint atomicMax; edge dedup uses a k*k
// bitmap with popcount-scan extraction (sorted-unique for free).
// ---------------------------------------------------------------------------

typedef float v2f_t __attribute__((ext_vector_type(2)));
typedef float v8f_t __attribute__((ext_vector_type(8)));

static constexpr int cN0    = 50000;
static constexpr int cE0    = 300000;
static constexpr int cK1    = 15000;
static constexpr int cK2    = 4500;
static constexpr int cDMAX0 = 32;
static constexpr int cDMAX1 = 64;
static constexpr int cE1CAP = 393216;
static constexpr int cE2CAP = 393216;
static constexpr int NPAD_TOP1 = 65536;   // pad for n=50000 sort
static constexpr int NPAD_TOP2 = 16384;   // pad for n=15000 sort
static constexpr int NPAD_EDGE = 524288;  // pad for (E+n) edge-order sort
static constexpr unsigned NEG_INF_ORD = 0x007FFFFFu; // f2o(-inf)

// ---- ordered float <-> uint (monotone) ------------------------------------
__device__ __forceinline__ unsigned f2o(float f) {
  unsigned u = __float_as_uint(f);
  return (u & 0x80000000u) ? ~u : (u | 0x80000000u);
}
__device__ __forceinline__ float o2f(unsigned u) {
  u = (u & 0x80000000u) ? (u & 0x7FFFFFFFu) : ~u;
  return __uint_as_float(u);
}

// ---- fills -----------------------------------------------------------------
__global__ void eun_fill_u32(unsigned* p, unsigned v, long n) {
  long i = (long)blockIdx.x * blockDim.x + threadIdx.x;
  if (i < n) p[i] = v;
}
__global__ void eun_fill_i32(int* p, int v, long n) {
  long i = (long)blockIdx.x * blockDim.x + threadIdx.x;
  if (i < n) p[i] = v;
}
__global__ void eun_fill_f32(float* p, float v, long n) {
  long i = (long)blockIdx.x * blockDim.x + threadIdx.x;
  if (i < n) p[i] = v;
}

// ---- W1 - W2 ---------------------------------------------------------------
__global__ void eun_wdiff(const float* __restrict__ W, float* __restrict__ wd,
                          int cin, int cout) {
  int i = blockIdx.x * blockDim.x + threadIdx.x;
  if (i < cin * cout) wd[i] = W[i] - W[cin * cout + i];
}

// ---- pack B[K x N] into WMMA fragment order --------------------------------
// Bp layout: [tc][ks][lane][j] with tc = N-tile, ks = K/4 step, lane = 0..31,
// j = 0..1.  Fragment value = B[ks*4 + 2*(lane>>4) + j][tc*16 + (lane&15)],
// zero-filled for col >= N.  One coalesced b64 load per lane per K-step.
__global__ void eun_pack_b(const float* __restrict__ B, float* __restrict__ Bp,
                           int K, int N) {
  int ntN = (N + 15) >> 4;
  int ks4 = K >> 2;
  long total = (long)ntN * ks4 * 64;
  long i = (long)blockIdx.x * blockDim.x + threadIdx.x;
  if (i >= total) return;
  int j    = (int)(i & 1);
  int lane = (int)((i >> 1) & 31);
  long g   = i >> 6;
  int ks   = (int)(g % ks4);
  int tc   = (int)(g / ks4);
  int h = lane >> 4, l = lane & 15;
  int row = ks * 4 + 2 * h + j;
  int col = tc * 16 + l;
  Bp[i] = (col < N) ? B[(long)row * N + col] : 0.f;
}

// ---- WMMA f32 GEMM: C[M x N] = A[M x K] @ B[K x N] (+bias) -----------------
// Block = 256 threads = 8 waves; each wave computes one 16x16 tile.
// Inner loop: global_load_b64 (A) + global_load_b64 (packed B) + v_wmma.
__global__ void eun_gemm_wmma(const float* __restrict__ A,
                              const float* __restrict__ Bp,
                              const float* __restrict__ bias,
                              float* __restrict__ C,
                              int M, int K, int N) {
  int wave = threadIdx.x >> 5;
  int lane = threadIdx.x & 31;
  int ntN  = (N + 15) >> 4;
  long tile = (long)blockIdx.x * 8 + wave;
  int tc = (int)(tile % ntN);
  int tm = (int)(tile / ntN) * 16;
  int tn = tc * 16;
  if (tm >= M) return;                       // uniform per wave: EXEC stays full
  int h = lane >> 4;                         // half-wave select
  int l = lane & 15;
  int rowA = tm + l;
  float amask = 1.f;
  if (rowA >= M) { rowA = M - 1; amask = 0.f; }  // clamp + 0/1 mask, no EXEC games
  const float* ap = A + (long)rowA * K + 2 * h;
  const float* bp = Bp + (long)tc * (K >> 2) * 64 + lane * 2;
  v8f_t acc = {};
  for (int kk = 0; kk < K; kk += 4) {
    v2f_t a, b;
    a[0] = ap[0] * amask;
    a[1] = ap[1] * amask;
    b[0] = bp[0];
    b[1] = bp[1];
    acc = __builtin_amdgcn_wmma_f32_16x16x4_f32(
        /*neg_a=*/false, a, /*neg_b=*/false, b,
        /*c_mod=*/(short)0, acc, /*reuse_a=*/false, /*reuse_b=*/false);
    ap += 4;
    bp += 64;
  }
  // C/D layout: VGPR r -> M = tm + r + 8*h, N = tn + (lane&15)
  for (int r = 0; r < 8; ++r) {
    int row = tm + r + 8 * h;
    int col = tn + l;
    if (row < M && col < N) {
      float v = acc[r];
      if (bias) v += bias[col];
      C[(long)row * N + col] = v;
    }
  }
}

// ---- edge scatter: agg[dst][c] = max(msg), deg[dst] += 1 -------------------
__global__ void eun_scatter_max(const float* __restrict__ Y,
                                const int* __restrict__ src,
                                const int* __restrict__ dst,
                                const int* __restrict__ valid,
                                int nE, int n, int cout,
                                unsigned* __restrict__ agg,
                                int* __restrict__ deg) {
  int e = blockIdx.x;
  if (e >= nE) return;
  if (valid && !valid[e]) return;
  int s = src[e]; if (s > n - 1) s = n - 1;   // srcc = min(src, n-1)
  int d = dst[e];
  int c = threadIdx.x;
  unsigned u = f2o(Y[(long)s * cout + c]);
  atomicMax(&agg[(long)d * cout + c], u);
  if (c == 0) atomicAdd(&deg[d], 1);
}

// ---- combine: out = (deg>0) ? maybe_relu(a + agg) : 0 ----------------------
__global__ void eun_combine(const float* __restrict__ a,
                            const unsigned* __restrict__ agg,
                            const int* __restrict__ deg,
                            float* __restrict__ out,
                            long total, int cout, int relu) {
  long idx = (long)blockIdx.x * blockDim.x + threadIdx.x;
  if (idx >= total) return;
  int i = (int)(idx / cout);
  float v = 0.f;
  if (deg[i] > 0) {
    v = a[idx] + o2f(agg[idx]);
    if (relu && v < 0.f) v = 0.f;
  }
  out[idx] = v;
}

// ---- top-k pooling ---------------------------------------------------------
__global__ void eun_norm(const float* __restrict__ w, int len, float* out) {
  __shared__ float sh[128];
  int t = threadIdx.x;
  float v = (t < len) ? w[t] : 0.f;
  sh[t] = v * v;
  __syncthreads();
  for (int off = 64; off > 0; off >>= 1) {
    if (t < off) sh[t] += sh[t + off];
    __syncthreads();
  }
  if (t == 0) out[0] = sqrtf(sh[0]);
}

__global__ void eun_score(const float* __restrict__ X, const float* __restrict__ w,
                          const float* __restrict__ nrm, float* __restrict__ score,
                          int n, int cin) {
  __shared__ float sw[128];
  if ((int)threadIdx.x < cin) sw[threadIdx.x] = w[threadIdx.x];
  __syncthreads();
  int i = blockIdx.x * blockDim.x + threadIdx.x;
  if (i >= n) return;
  const float* row = X + (long)i * cin;
  float acc = 0.f;
  for (int c = 0; c < cin; ++c) acc += row[c] * sw[c];
  score[i] = acc / nrm[0];
}

// key = (~ordered(score)) << 32 | index  -> ascending sort gives top_k order
__global__ void eun_build_topk_keys(const float* __restrict__ score,
                                    unsigned long long* __restrict__ keys,
                                    int n, int npad) {
  int i = blockIdx.x * blockDim.x + threadIdx.x;
  if (i >= npad) return;
  if (i < n)
    keys[i] = (((unsigned long long)(~f2o(score[i]))) << 32) | (unsigned)i;
  else
    keys[i] = 0xFFFFFFFFFFFFFFFFull;
}

__global__ void eun_extract_topk(const unsigned long long* __restrict__ keys,
                                 const float* __restrict__ score,
                                 const float* __restrict__ Xin,
                                 float* __restrict__ Xout,
                                 int* __restrict__ perm, int cin) {
  int i = blockIdx.x;
  unsigned idx = (unsigned)(keys[i] & 0xFFFFFFFFull);
  if (threadIdx.x == 0) perm[i] = (int)idx;
  float t = tanhf(score[idx]);
  for (int c = threadIdx.x; c < cin; c += blockDim.x)
    Xout[(long)i * cin + c] = Xin[(long)idx * cin + c] * t;
}

__global__ void eun_inv_set(const int* __restrict__ perm, int* __restrict__ inv, int k) {
  int i = blockIdx.x * blockDim.x + threadIdx.x;
  if (i < k) inv[perm[i]] = i;
}

// ---- bitonic sort (u64, n = power of two) ----------------------------------
__global__ void eun_bitonic_step(unsigned long long* __restrict__ d,
                                 unsigned j, unsigned k) {
  unsigned i = blockIdx.x * blockDim.x + threadIdx.x;
  unsigned ixj = i ^ j;
  if (ixj > i) {
    unsigned long long a = d[i], b = d[ixj];
    bool asc = ((i & k) == 0);
    if ((a > b) == asc) { d[i] = b; d[ixj] = a; }
  }
}

// ---- edge augmentation -----------------------------------------------------
// combined arrays over ea = E + n entries (original edges + self loops)
__global__ void eun_build_comb(const int* __restrict__ src, const int* __restrict__ dst,
                               const int* __restrict__ valid, int E, int n,
                               int* __restrict__ sRaw, int* __restrict__ cC,
                               int* __restrict__ vC) {
  int i = blockIdx.x * blockDim.x + threadIdx.x;
  int ea = E + n;
  if (i >= ea) return;
  int s, d, v;
  if (i < E) {
    s = src[i]; d = dst[i];
    int bv = valid ? valid[i] : 1;
    v = (bv && (s != d)) ? 1 : 0;
  } else {
    s = i - E; d = s; v = 1;
  }
  sRaw[i] = s;
  vC[i]   = v;
  cC[i]   = v ? d : n;
}

__global__ void eun_build_rkeys(const int* __restrict__ sRaw, const int* __restrict__ vC,
                                unsigned long long* __restrict__ keys,
                                int ea, int npad, int n) {
  int i = blockIdx.x * blockDim.x + threadIdx.x;
  if (i >= npad) return;
  if (i < ea) {
    int r = vC[i] ? sRaw[i] : n;
    keys[i] = (((unsigned long long)(unsigned)r) << 32) | (unsigned)i;
  } else {
    keys[i] = 0xFFFFFFFFFFFFFFFFull;
  }
}

// nbr[r][slot] = c for first dmax edges of each row (original order preserved)
__global__ void eun_nbr_build(const unsigned long long* __restrict__ keys,
                              const int* __restrict__ cC, int* __restrict__ nbr,
                              int ea, int n, int dmax) {
  int p = blockIdx.x * blockDim.x + threadIdx.x;
  if (p >= ea) return;
  unsigned long long key = keys[p];
  int r = (int)(key >> 32);
  int i = (int)(key & 0xFFFFFFFFull);
  unsigned long long tgt = ((unsigned long long)(unsigned)r) << 32;
  int lo = 0, hi = ea;
  while (lo < hi) {
    int mid = (lo + hi) >> 1;
    if (keys[mid] < tgt) lo = mid + 1; else hi = mid;
  }
  int slot = p - lo;
  if (slot < dmax) nbr[r * dmax + slot] = cC[i];
}

// set bitmap bit nr*k+nc for every valid 2-hop edge
__global__ void eun_populate(const int* __restrict__ sRaw, const int* __restrict__ cC,
                             const int* __restrict__ vC, const int* __restrict__ nbr,
                             const int* __restrict__ inv, int n, int kdim, int dmax,
                             int ea, unsigned* __restrict__ bm) {
  int idx = blockIdx.x * blockDim.x + threadIdx.x;
  if (idx >= ea * dmax) return;
  int j = idx / dmax;
  int t = idx - j * dmax;
  if (!vC[j]) return;
  int c    = cC[j];
  int col2 = nbr[c * dmax + t];
  int row2 = sRaw[j];
  if (col2 >= n || row2 == col2) return;
  int r2 = (row2 < n - 1) ? row2 : (n - 1);
  int nr = inv[r2];
  int nc = inv[col2];
  if (nr >= kdim || nc >= kdim) return;
  int key = nr * kdim + nc;
  atomicOr(&bm[key >> 5], 1u << (key & 31));
}

__global__ void eun_count_words(const unsigned* __restrict__ bm,
                                unsigned* __restrict__ bcnt, int nwords) {
  __shared__ unsigned sh[512];
  int t = threadIdx.x;
  int wi = blockIdx.x * 512 + t;
  sh[t] = (wi < nwords) ? (unsigned)__popc(bm[wi]) : 0u;
  __syncthreads();
  for (int off = 256; off > 0; off >>= 1) {
    if (t < off) sh[t] += sh[t + off];
    __syncthreads();
  }
  if (t == 0) bcnt[blockIdx.x] = sh[0];
}

__global__ void eun_exscan(const unsigned* __restrict__ in,
                           unsigned* __restrict__ out, int n) {
  __shared__ unsigned shA[1024];
  __shared__ unsigned shB[1024];
  int t = threadIdx.x;
  int chunk = (n + 1023) >> 10;
  int lo = t * chunk;
  int hi = lo + chunk; if (hi > n) hi = n;
  unsigned s = 0;
  for (int i = lo; i < hi; ++i) s += in[i];
  shA[t] = s;
  __syncthreads();
  unsigned* a = shA; unsigned* b = shB;
  for (int off = 1; off < 1024; off <<= 1) {
    unsigned v = a[t];
    if (t >= off) v += a[t - off];
    b[t] = v;
    __syncthreads();
    unsigned* tmp = a; a = b; b = tmp;
  }
  unsigned base = (t == 0) ? 0u : a[t - 1];
  for (int i = lo; i < hi; ++i) { unsigned v = in[i]; out[i] = base; base += v; }
}

// extract set bits (globally ascending) -> (so, do, vo) for pos < ecap
__global__ void eun_emit(const unsigned* __restrict__ bm,
                         const unsigned* __restrict__ boff,
                         int nwords, int kdim, int ecap,
                         int* __restrict__ so, int* __restrict__ dO,
                         int* __restrict__ vO) {
  __shared__ unsigned shA[512];
  __shared__ unsigned shB[512];
  int t = threadIdx.x;
  int wi = blockIdx.x * 512 + t;
  unsigned w = (wi < nwords) ? bm[wi] : 0u;
  unsigned pc = (unsigned)__popc(w);
  shA[t] = pc;
  __syncthreads();
  unsigned* a = shA; unsigned* b = shB;
  for (int off = 1; off < 512; off <<= 1) {
    unsigned v = a[t];
    if (t >= off) v += a[t - off];
    b[t] = v;
    __syncthreads();
    unsigned* tmp = a; a = b; b = tmp;
  }
  unsigned base = boff[blockIdx.x] + a[t] - pc;
  while (w) {
    int bit = __ffs((int)w) - 1;
    w &= (w - 1u);
    int pos = (int)base; base++;
    if (pos < ecap) {
      int key = wi * 32 + bit;
      so[pos] = key / kdim;
      dO[pos] = key % kdim;
      vO[pos] = 1;
    }
  }
}

// ---- concat helpers --------------------------------------------------------
__global__ void eun_pack(const float* __restrict__ X, float* __restrict__ dst,
                         int rows, int cols, int stride, int off) {
  long idx = (long)blockIdx.x * blockDim.x + threadIdx.x;
  if (idx >= (long)rows * cols) return;
  int r = (int)(idx / cols), c = (int)(idx % cols);
  dst[(long)r * stride + off + c] = X[idx];
}
__global__ void eun_scatter_rows(const float* __restrict__ X,
                                 const int* __restrict__ perm,
                                 float* __restrict__ dst,
                                 int rows, int cols, int stride, int off) {
  long idx = (long)blockIdx.x * blockDim.x + threadIdx.x;
  if (idx >= (long)rows * cols) return;
  int r = (int)(idx / cols), c = (int)(idx % cols);
  dst[(long)perm[r] * stride + off + c] = X[idx];
}

// ---------------------------------------------------------------------------
static void sort_u64(unsigned long long* d, unsigned n, hipStream_t stream) {
  dim3 blk(256), grd(n / 256);
  for (unsigned k = 2; k <= n; k <<= 1)
    for (unsigned j = k >> 1; j > 0; j >>= 1)
      eun_bitonic_step<<<grd, blk, 0, stream>>>(d, j, k);
}

extern "C" void kernel_launch(void* const* d_in, const int* in_sizes, int n_in,
                              void* d_out, int out_size, void* d_ws, size_t ws_size,
                              hipStream_t stream) {
  (void)in_sizes; (void)n_in; (void)out_size; (void)ws_size;

  const float* x       = (const float*)d_in[0];
  const int*   ei      = (const int*)  d_in[1];
  const float* W_d0    = (const float*)d_in[2];
  const float* b_d0    = (const float*)d_in[3];
  const float* pool_w0 = (const float*)d_in[4];
  const float* W_d1    = (const float*)d_in[5];
  const float* b_d1    = (const float*)d_in[6];
  const float* pool_w1 = (const float*)d_in[7];
  const float* W_d2    = (const float*)d_in[8];
  const float* b_d2    = (const float*)d_in[9];
  const float* W_u0    = (const float*)d_in[10];
  const float* b_u0    = (const float*)d_in[11];
  const float* W_u1    = (const float*)d_in[12];
  const float* b_u1    = (const float*)d_in[13];
  const float* W_dec   = (const float*)d_in[14];
  const float* b_dec   = (const float*)d_in[15];
  const int* src0 = ei;
  const int* dst0 = ei + cE0;
  float* out = (float*)d_out;

  // ---- workspace arena -----------------------------------------------------
  char* base = (char*)d_ws;
  size_t off = 0;
  auto alloc = [&](size_t bytes) -> void* {
    void* p = base + off;
    off = (off + bytes + 255) & ~(size_t)255;
    return p;
  };
  float* x0   = (float*)alloc((size_t)cN0 * 128 * 4);
  float* x1   = (float*)alloc((size_t)cK1 * 128 * 4);
  float* x2   = (float*)alloc((size_t)cK2 * 128 * 4);
  float* x1p  = (float*)alloc((size_t)cK1 * 128 * 4);
  float* x2p  = (float*)alloc((size_t)cK2 * 128 * 4);
  float* h1   = (float*)alloc((size_t)cK1 * 128 * 4);
  float* h0   = (float*)alloc((size_t)cN0 * 64 * 4);
  float* gA   = (float*)alloc((size_t)cN0 * 128 * 4);
  float* gY   = (float*)alloc((size_t)cN0 * 128 * 4);
  unsigned* agg = (unsigned*)alloc((size_t)cN0 * 128 * 4);
  int*   deg  = (int*)alloc((size_t)cN0 * 4);
  float* wdiff = (float*)alloc((size_t)256 * 128 * 4);
  float* bpack = (float*)alloc((size_t)8 * 64 * 64 * 4);    // ntN<=8, K/4<=64
  float* score = (float*)alloc((size_t)cN0 * 4);
  float* normbuf = (float*)alloc(256);
  unsigned long long* sortbuf = (unsigned long long*)alloc((size_t)NPAD_EDGE * 8);
  int* sRaw = (int*)alloc((size_t)NPAD_EDGE * 4);
  int* cC   = (int*)alloc((size_t)NPAD_EDGE * 4);
  int* vC   = (int*)alloc((size_t)NPAD_EDGE * 4);
  int* nbr  = (int*)alloc((size_t)(cN0 + 1) * cDMAX0 * 4);   // >= (K1+1)*DMAX1
  unsigned* bitmap = (unsigned*)alloc((size_t)7031250 * 4);  // ceil(15000^2/32)
  unsigned* bcnt = (unsigned*)alloc((size_t)16384 * 4);
  unsigned* boff = (unsigned*)alloc((size_t)16384 * 4);
  int* perm1 = (int*)alloc((size_t)cK1 * 4);
  int* perm2 = (int*)alloc((size_t)cK2 * 4);
  int* inv1  = (int*)alloc((size_t)cN0 * 4);
  int* inv2  = (int*)alloc((size_t)cK1 * 4);
  int* s1 = (int*)alloc((size_t)cE1CAP * 4);
  int* d1 = (int*)alloc((size_t)cE1CAP * 4);
  int* v1 = (int*)alloc((size_t)cE1CAP * 4);
  int* s2 = (int*)alloc((size_t)cE2CAP * 4);
  int* d2 = (int*)alloc((size_t)cE2CAP * 4);
  int* v2 = (int*)alloc((size_t)cE2CAP * 4);
  float* concat = (float*)alloc((size_t)cN0 * 256 * 4);

  auto gb = [](long n) { return (unsigned)((n + 255) / 256); };

  // ---- helpers -------------------------------------------------------------
  auto gemm = [&](const float* A, const float* B, const float* bias, float* C,
                  int M, int K, int N) {
    int ntN = (N + 15) / 16;
    long btot = (long)ntN * (K / 4) * 64;
    eun_pack_b<<<gb(btot), 256, 0, stream>>>(B, bpack, K, N);
    long tiles = (long)((M + 15) / 16) * ntN;
    eun_gemm_wmma<<<(unsigned)((tiles + 7) / 8), 256, 0, stream>>>(A, bpack, bias, C, M, K, N);
  };

  auto edge_conv = [&](const float* X, int cin, const float* W, const float* b,
                       int cout, const int* src, const int* dst, const int* valid,
                       int nE, int n, float* outp, int relu) {
    eun_wdiff<<<gb((long)cin * cout), 256, 0, stream>>>(W, wdiff, cin, cout);
    gemm(X, wdiff, b, gA, n, cin, cout);                 // a = X@(W1-W2)+b
    gemm(X, W + (size_t)cin * cout, nullptr, gY, n, cin, cout); // y = X@W2
    eun_fill_u32<<<gb((long)n * cout), 256, 0, stream>>>(agg, NEG_INF_ORD, (long)n * cout);
    eun_fill_i32<<<gb(n), 256, 0, stream>>>(deg, 0, n);
    eun_scatter_max<<<(unsigned)nE, (unsigned)cout, 0, stream>>>(
        gY, src, dst, valid, nE, n, cout, agg, deg);
    long tot = (long)n * cout;
    eun_combine<<<gb(tot), 256, 0, stream>>>(gA, agg, deg, outp, tot, cout, relu);
  };

  auto topk = [&](const float* Xin, int n, const float* w, int kdim, int npad,
                  float* Xout, int* perm, int* inv, int invLen) {
    eun_norm<<<1, 128, 0, stream>>>(w, 128, normbuf);
    eun_score<<<gb(n), 256, 0, stream>>>(Xin, w, normbuf, score, n, 128);
    eun_build_topk_keys<<<(unsigned)(npad / 256), 256, 0, stream>>>(score, sortbuf, n, npad);
    sort_u64(sortbuf, (unsigned)npad, stream);
    eun_extract_topk<<<(unsigned)kdim, 128, 0, stream>>>(sortbuf, score, Xin, Xout, perm, 128);
    eun_fill_i32<<<gb(invLen), 256, 0, stream>>>(inv, kdim, invLen);
    eun_inv_set<<<gb(kdim), 256, 0, stream>>>(perm, inv, kdim);
  };

  auto augment = [&](const int* src, const int* dst, const int* valid, int E,
                     int n, int kdim, int dmax, const int* inv, int ecap,
                     int* so, int* dO, int* vO) {
    int ea = E + n;
    eun_build_comb<<<gb(ea), 256, 0, stream>>>(src, dst, valid, E, n, sRaw, cC, vC);
    eun_build_rkeys<<<(unsigned)(NPAD_EDGE / 256), 256, 0, stream>>>(
        sRaw, vC, sortbuf, ea, NPAD_EDGE, n);
    sort_u64(sortbuf, (unsigned)NPAD_EDGE, stream);
    eun_fill_i32<<<gb((long)(n + 1) * dmax), 256, 0, stream>>>(nbr, n, (long)(n + 1) * dmax);
    eun_nbr_build<<<gb(ea), 256, 0, stream>>>(sortbuf, cC, nbr, ea, n, dmax);
    long nwords = ((long)kdim * kdim + 31) / 32;
    eun_fill_u32<<<gb(nwords), 256, 0, stream>>>(bitmap, 0u, nwords);
    eun_populate<<<gb((long)ea * dmax), 256, 0, stream>>>(
        sRaw, cC, vC, nbr, inv, n, kdim, dmax, ea, bitmap);
    int nb = (int)((nwords + 511) / 512);
    eun_count_words<<<(unsigned)nb, 512, 0, stream>>>(bitmap, bcnt, (int)nwords);
    eun_exscan<<<1, 1024, 0, stream>>>(bcnt, boff, nb);
    eun_fill_i32<<<gb(ecap), 256, 0, stream>>>(so, 0, ecap);
    eun_fill_i32<<<gb(ecap), 256, 0, stream>>>(dO, 0, ecap);
    eun_fill_i32<<<gb(ecap), 256, 0, stream>>>(vO, 0, ecap);
    eun_emit<<<(unsigned)nb, 512, 0, stream>>>(bitmap, boff, (int)nwords, kdim,
                                               ecap, so, dO, vO);
  };

  // ---- pipeline ------------------------------------------------------------
  edge_conv(x, 32, W_d0, b_d0, 128, src0, dst0, nullptr, cE0, cN0, x0, 1);
  topk(x0, cN0, pool_w0, cK1, NPAD_TOP1, x1p, perm1, inv1, cN0);
  augment(src0, dst0, nullptr, cE0, cN0, cK1, cDMAX0, inv1, cE1CAP, s1, d1, v1);

  edge_conv(x1p, 128, W_d1, b_d1, 128, s1, d1, v1, cE1CAP, cK1, x1, 1);
  topk(x1, cK1, pool_w1, cK2, NPAD_TOP2, x2p, perm2, inv2, cK1);
  augment(s1, d1, v1, cE1CAP, cK1, cK2, cDMAX1, inv2, cE2CAP, s2, d2, v2);

  edge_conv(x2p, 128, W_d2, b_d2, 128, s2, d2, v2, cE2CAP, cK2, x2, 1);

  // up path 1: concat1 = [x1, scatter(perm2, x2)]  (K1 x 256)
  eun_fill_f32<<<gb((long)cK1 * 256), 256, 0, stream>>>(concat, 0.f, (long)cK1 * 256);
  eun_pack<<<gb((long)cK1 * 128), 256, 0, stream>>>(x1, concat, cK1, 128, 256, 0);
  eun_scatter_rows<<<gb((long)cK2 * 128), 256, 0, stream>>>(x2, perm2, concat, cK2, 128, 256, 128);
  edge_conv(concat, 256, W_u0, b_u0, 128, s1, d1, v1, cE1CAP, cK1, h1, 1);

  // up path 0: concat0 = [x0, scatter(perm1, h1)]  (N0 x 256)
  eun_fill_f32<<<gb((long)cN0 * 256), 256, 0, stream>>>(concat, 0.f, (long)cN0 * 256);
  eun_pack<<<gb((long)cN0 * 128), 256, 0, stream>>>(x0, concat, cN0, 128, 256, 0);
  eun_scatter_rows<<<gb((long)cK1 * 128), 256, 0, stream>>>(h1, perm1, concat, cK1, 128, 256, 128);
  edge_conv(concat, 256, W_u1, b_u1, 64, src0, dst0, nullptr, cE0, cN0, h0, 0);

  // decoder: out = h0 @ W_dec + b_dec  (N0 x 8)
  gemm(h0, W_dec, b_dec, out, cN0, 64, 8);
}